// social_stgcn_70179765616973
// MI455X (gfx1250) — compile-verified
//
#include <hip/hip_runtime.h>
#include <math.h>

typedef float v2f __attribute__((ext_vector_type(2)));
typedef float v8f __attribute__((ext_vector_type(8)));

// ---------------------------------------------------------------------------
// Graph side: degree, first (2->16) matmul, edge scatter-add, finalize.
// ---------------------------------------------------------------------------

__global__ void k_fill_deg(float* __restrict__ deg, int N) {
  int i = blockIdx.x * blockDim.x + threadIdx.x;
  if (i < N) deg[i] = 2.0f;                       // improved self-loop weight 2
}

__global__ void k_count_deg(const int* __restrict__ col, float* __restrict__ deg, int E) {
  int e = blockIdx.x * blockDim.x + threadIdx.x;
  if (e < E) atomicAdd(&deg[col[e]], 1.0f);
}

__global__ void k_gcn1_mm(const float* __restrict__ x, const float* __restrict__ W,
                          float* __restrict__ xw, int N) {
  int i = blockIdx.x * blockDim.x + threadIdx.x;
  if (i >= N) return;
  float x0 = x[2 * i], x1 = x[2 * i + 1];
#pragma unroll
  for (int j = 0; j < 16; ++j)
    xw[16 * i + j] = fmaf(x0, W[j], x1 * W[16 + j]);
}

__global__ void k_zero(float* __restrict__ p, unsigned n) {
  unsigned i = blockIdx.x * blockDim.x + threadIdx.x;
  if (i < n) p[i] = 0.0f;
}

// 16 lanes per edge: lane j scatters feature j. xw row + agg row are each one
// contiguous 64B segment -> coalesced gather and coalesced f32 atomics.
__global__ void k_aggregate(const int* __restrict__ row, const int* __restrict__ col,
                            const float* __restrict__ deg, const float* __restrict__ xw,
                            float* __restrict__ agg, unsigned E) {
  unsigned t = blockIdx.x * blockDim.x + threadIdx.x;
  unsigned e = t >> 4;
  unsigned j = t & 15u;
  if (e >= E) return;
  int r = row[e], c = col[e];
  float norm = rsqrtf(deg[r]) * rsqrtf(deg[c]);
  atomicAdd(&agg[(unsigned)c * 16u + j], xw[(unsigned)r * 16u + j] * norm);
}

// h = relu(agg + (2/deg)*xw + b); safe to run with h aliasing agg.
__global__ void k_gcn_finalize(const float* __restrict__ agg, const float* __restrict__ xw,
                               const float* __restrict__ deg, const float* __restrict__ b,
                               float* __restrict__ h, unsigned n16) {
  unsigned t = blockIdx.x * blockDim.x + threadIdx.x;
  if (t >= n16) return;
  unsigned i = t >> 4, j = t & 15u;
  float v = agg[t] + (2.0f / deg[i]) * xw[t] + b[j];
  h[t] = fmaxf(v, 0.0f);
}

// ---------------------------------------------------------------------------
// WMMA helpers: one wave owns a 16-node tile; K=16 via 4x V_WMMA_F32_16X16X4_F32.
// A layout (16x4 f32): lane l -> (M=l&15, K=2*(l>>4)+j)
// B layout (4x16 f32): lane l -> (K=2*(l>>4)+j, N=l&15)
// C/D (v8f):           c[r]   -> (M=r+8*(l>>4), N=l&15)
// ---------------------------------------------------------------------------

__device__ __forceinline__ void load_a_tile(const float* __restrict__ in, unsigned base,
                                            unsigned m, unsigned hi, unsigned N, v2f a[4]) {
  unsigned rowi = base + m;
  if (rowi >= N) rowi = N - 1;                    // tail clamp (N%16==0 in practice)
  const float* p = in + (size_t)rowi * 16u + 2u * hi;
#pragma unroll
  for (int s = 0; s < 4; ++s) {
    a[s].x = p[4 * s];
    a[s].y = p[4 * s + 1];
  }
}

__device__ __forceinline__ v8f mm16(const v2f a[4], const float* __restrict__ W,
                                    unsigned hi, unsigned n, v8f c) {
#pragma unroll
  for (int s = 0; s < 4; ++s) {
    v2f b;
    unsigned k = 4u * (unsigned)s + 2u * hi;
    b.x = W[k * 16u + n];
    b.y = W[(k + 1u) * 16u + n];
    c = __builtin_amdgcn_wmma_f32_16x16x4_f32(false, a[s], false, b, (short)0, c,
                                              false, false);
  }
  return c;
}

// Branch-free tile store for full tiles: one base pointer + immediate offsets.
__device__ __forceinline__ void store_tile(float* __restrict__ out, unsigned base,
                                           unsigned m, unsigned hi, unsigned N,
                                           const v8f& c) {
  if (base + 16u <= N) {
    float* p = out + ((size_t)(base + 8u * hi) * 16u + m);
#pragma unroll
    for (int r = 0; r < 8; ++r) p[(size_t)r * 16u] = c[r];
  } else {
#pragma unroll
    for (int r = 0; r < 8; ++r) {
      unsigned rowi = base + (unsigned)r + 8u * hi;
      if (rowi < N) out[(size_t)rowi * 16u + m] = c[r];
    }
  }
}

__device__ __forceinline__ float fast_sigmoid(float x) {
  return __frcp_rn(1.0f + __expf(-x));
}
__device__ __forceinline__ float fast_tanh(float x) {
  // tanh(x) = 1 - 2/(exp(2x)+1)
  return fmaf(-2.0f, __frcp_rn(__expf(2.0f * x) + 1.0f), 1.0f);
}

// out[N,16] = in[N,16] @ W[16,16]   (no bias / activation: GCN2 pre-aggregation)
__global__ void k_wmma_gemm16(const float* __restrict__ in, const float* __restrict__ W,
                              float* __restrict__ out, unsigned N) {
  unsigned wave = (blockIdx.x * blockDim.x + threadIdx.x) >> 5;
  unsigned lane = threadIdx.x & 31u;
  unsigned base = wave * 16u;
  if (base >= N) return;
  unsigned m = lane & 15u, hi = lane >> 4;
  v2f a[4];
  load_a_tile(in, base, m, hi, N, a);
  v8f c = {};
  c = mm16(a, W, hi, m, c);
  store_tile(out, base, m, hi, N, c);
}

// Fused GConvLSTM step (zero initial H/C): 3 matmuls sharing one A tile,
// then per-lane gate math. out_c=0 -> relu(H), out_c=1 -> relu(C).
__global__ void k_lstm_fused(const float* __restrict__ hin,
                             const float* __restrict__ wi, const float* __restrict__ wc,
                             const float* __restrict__ wo,
                             const float* __restrict__ bi, const float* __restrict__ bc,
                             const float* __restrict__ bo, const float* __restrict__ wco,
                             float* __restrict__ outp, unsigned N, int out_c) {
  unsigned wave = (blockIdx.x * blockDim.x + threadIdx.x) >> 5;
  unsigned lane = threadIdx.x & 31u;
  unsigned base = wave * 16u;
  if (base >= N) return;
  unsigned m = lane & 15u, hi = lane >> 4;

  v2f a[4];
  load_a_tile(hin, base, m, hi, N, a);

  float Bi = bi[m], Bc = bc[m], Bo = bo[m], Wc = wco[m];
  v8f ci, cc, co;
#pragma unroll
  for (int r = 0; r < 8; ++r) { ci[r] = Bi; cc[r] = Bc; co[r] = Bo; }

  ci = mm16(a, wi, hi, m, ci);
  cc = mm16(a, wc, hi, m, cc);
  co = mm16(a, wo, hi, m, co);

  v8f res;
#pragma unroll
  for (int r = 0; r < 8; ++r) {
    float ig = fast_sigmoid(ci[r]);
    float C  = ig * fast_tanh(cc[r]);
    float v;
    if (out_c) {
      v = C;
    } else {
      float og = fast_sigmoid(co[r] + Wc * C);
      v = og * fast_tanh(C);
    }
    res[r] = fmaxf(v, 0.0f);
  }
  store_tile(outp, base, m, hi, N, res);
}

__global__ void k_final_lin(const float* __restrict__ c, const float* __restrict__ lw,
                            const float* __restrict__ lb, float* __restrict__ out, int N) {
  int i = blockIdx.x * blockDim.x + threadIdx.x;
  if (i >= N) return;
  float a0 = lb[0], a1 = lb[1], a2 = lb[2];
#pragma unroll
  for (int j = 0; j < 16; ++j) {
    float v = c[16 * i + j];
    a0 = fmaf(v, lw[3 * j + 0], a0);
    a1 = fmaf(v, lw[3 * j + 1], a1);
    a2 = fmaf(v, lw[3 * j + 2], a2);
  }
  out[3 * i + 0] = a0;
  out[3 * i + 1] = a1;
  out[3 * i + 2] = a2;
}

// ---------------------------------------------------------------------------

extern "C" void kernel_launch(void* const* d_in, const int* in_sizes, int n_in,
                              void* d_out, int out_size, void* d_ws, size_t ws_size,
                              hipStream_t stream) {
  (void)n_in; (void)out_size; (void)ws_size;
  const int N = in_sizes[0] / 2;
  const int E = in_sizes[1] / 2;

  const float* x   = (const float*)d_in[0];
  const int*   ei  = (const int*)d_in[1];
  const int*   row = ei;
  const int*   col = ei + E;
  const float* g1w = (const float*)d_in[2];
  const float* g1b = (const float*)d_in[3];
  const float* g2w = (const float*)d_in[4];
  const float* g2b = (const float*)d_in[5];
  const float* lw  = (const float*)d_in[6];
  const float* lb  = (const float*)d_in[7];
  const float* L[3][7];
  for (int l = 0; l < 3; ++l)
    for (int k = 0; k < 7; ++k)
      L[l][k] = (const float*)d_in[8 + 7 * l + k];   // wi,wc,wo,bi,bc,bo,wco

  float* ws   = (float*)d_ws;
  float* deg  = ws;                                  // [N]
  float* bufA = ws + N;                              // [N,16]
  float* bufB = ws + N + (size_t)N * 16;             // [N,16]

  const int B = 256;
  const unsigned n16 = (unsigned)N * 16u;
  const unsigned nodeBlocks = (N + B - 1) / B;
  const unsigned featBlocks = (n16 + B - 1) / B;
  const unsigned edgeBlocks = ((unsigned)E + B - 1) / B;
  const unsigned edgeFeatBlocks = ((unsigned)E * 16u + B - 1) / B;
  const unsigned tiles = ((unsigned)N + 15u) / 16u;
  const unsigned waveBlocks = (tiles * 32u + B - 1) / B;

  // degrees (shared by both GCN layers)
  k_fill_deg<<<nodeBlocks, B, 0, stream>>>(deg, N);
  k_count_deg<<<edgeBlocks, B, 0, stream>>>(col, deg, E);

  // ---- GCN layer 1 ----
  k_gcn1_mm<<<nodeBlocks, B, 0, stream>>>(x, g1w, bufA, N);          // xw1 -> A
  k_zero<<<featBlocks, B, 0, stream>>>(bufB, n16);
  k_aggregate<<<edgeFeatBlocks, B, 0, stream>>>(row, col, deg, bufA, bufB, (unsigned)E);
  k_gcn_finalize<<<featBlocks, B, 0, stream>>>(bufB, bufA, deg, g1b, bufB, n16); // h1 -> B

  // ---- GCN layer 2 ----
  k_wmma_gemm16<<<waveBlocks, B, 0, stream>>>(bufB, g2w, bufA, (unsigned)N);     // xw2 -> A
  k_zero<<<featBlocks, B, 0, stream>>>(bufB, n16);
  k_aggregate<<<edgeFeatBlocks, B, 0, stream>>>(row, col, deg, bufA, bufB, (unsigned)E);
  k_gcn_finalize<<<featBlocks, B, 0, stream>>>(bufB, bufA, deg, g2b, bufB, n16); // h2 -> B

  // ---- GConvLSTM stack (fused, WMMA) ----
  k_lstm_fused<<<waveBlocks, B, 0, stream>>>(bufB, L[0][0], L[0][1], L[0][2],
                                             L[0][3], L[0][4], L[0][5], L[0][6],
                                             bufA, (unsigned)N, 0);              // H1 -> A
  k_lstm_fused<<<waveBlocks, B, 0, stream>>>(bufA, L[1][0], L[1][1], L[1][2],
                                             L[1][3], L[1][4], L[1][5], L[1][6],
                                             bufB, (unsigned)N, 0);              // H2 -> B
  k_lstm_fused<<<waveBlocks, B, 0, stream>>>(bufB, L[2][0], L[2][1], L[2][2],
                                             L[2][3], L[2][4], L[2][5], L[2][6],
                                             bufA, (unsigned)N, 1);              // C3 -> A

  // ---- output projection ----
  k_final_lin<<<nodeBlocks, B, 0, stream>>>(bufA, lw, lb, (float*)d_out, N);
}